// KGAT_rotatE_55241869361805
// MI455X (gfx1250) — compile-verified
//
#include <hip/hip_runtime.h>
#include <hip/hip_bf16.h>
#include <math.h>

#define NNODE 100000
#define NEDGE 1000000
#define DHALF 64

typedef __attribute__((ext_vector_type(2))) float v2f;
typedef __attribute__((ext_vector_type(8))) float v8f;

// ---------------------------------------------------------------- utilities
__global__ void zero_kernel(float* p, long long n) {
    long long i = (long long)blockIdx.x * blockDim.x + threadIdx.x;
    if (i < n) p[i] = 0.0f;
}

__global__ void copy_embed_kernel(const float* __restrict__ ent, float* __restrict__ out,
                                  long long total) {
    long long i = (long long)blockIdx.x * blockDim.x + threadIdx.x;
    if (i >= total) return;
    long long n = i >> 7;          // /128
    int j = (int)(i & 127);
    out[n * 240 + j] = ent[i];
}

// ------------------------------------------------- phase 1: edge attention
// one wave per edge; lane l covers dims l and l+32 of D=64
__global__ void edge_att_kernel(const float* __restrict__ ent, const float* __restrict__ rel,
                                const int* __restrict__ src, const int* __restrict__ dst,
                                const int* __restrict__ typ, float* __restrict__ att,
                                unsigned int* __restrict__ segmax, float phase_scale, int E) {
    int wid = (blockIdx.x * blockDim.x + threadIdx.x) >> 5;
    if (wid >= E) return;
    int lane = threadIdx.x & 31;
    int s = src[wid], d = dst[wid], r = typ[wid];
    const float* h  = ent + (size_t)s * 128;
    const float* tt = ent + (size_t)d * 128;
    const float* rr = rel + (size_t)r * DHALF;
    float sum = 0.0f;
#pragma unroll
    for (int q = 0; q < 2; ++q) {
        int j = lane + q * 32;
        float reh = h[j],  imh = h[j + DHALF];
        float ret = tt[j], imt = tt[j + DHALF];
        float ph = rr[j] * phase_scale;
        float sr, cr;
        sincosf(ph, &sr, &cr);
        float rs = reh * cr - imh * sr - ret;
        float is = reh * sr + imh * cr - imt;
        sum += sqrtf(rs * rs + is * is);
    }
#pragma unroll
    for (int off = 16; off > 0; off >>= 1) sum += __shfl_xor(sum, off, 32);
    if (lane == 0) {
        att[wid] = sum;
        // sum >= 0 always, so uint compare order == float compare order
        atomicMax(segmax + d, __float_as_uint(sum));
    }
}

__global__ void edge_expsum_kernel(const float* __restrict__ segmax, float* __restrict__ att,
                                   const int* __restrict__ dst, float* __restrict__ segsum, int E) {
    int e = blockIdx.x * blockDim.x + threadIdx.x;
    if (e >= E) return;
    int d = dst[e];
    float ex = expf(att[e] - segmax[d]);
    att[e] = ex;
    atomicAdd(segsum + d, ex);
}

__global__ void edge_norm_kernel(float* __restrict__ att, const int* __restrict__ dst,
                                 const float* __restrict__ segsum, int E) {
    int e = blockIdx.x * blockDim.x + threadIdx.x;
    if (e >= E) return;
    att[e] = att[e] / segsum[dst[e]];
}

// ------------------------------------------------- per-layer message scatter
// one wave per edge, lanes stride the feature dim; atomic f32 adds
template <int DIN>
__global__ void scatter_kernel(const float* __restrict__ ego, const float* __restrict__ att,
                               const int* __restrict__ src, const int* __restrict__ dst,
                               float* __restrict__ side, int E) {
    int wid = (blockIdx.x * blockDim.x + threadIdx.x) >> 5;
    if (wid >= E) return;
    int lane = threadIdx.x & 31;
    int s = src[wid], d = dst[wid];
    float a = att[wid];
    const float* eg = ego + (size_t)s * DIN;
    float* sd = side + (size_t)d * DIN;
#pragma unroll
    for (int j = lane; j < DIN; j += 32) atomicAdd(sd + j, a * eg[j]);
}

// ------------------------------------------------- node dual-MLP via WMMA f32
// wave owns a 16-node tile.  A-frag: lane = (half, mrow): holds X[m0+mrow][k+2*half .. +1]
// B-frag: holds W[k+2*half .. +1][ncol0+mrow].  C: vgpr i, row = m0 + i + 8*half, col = mrow.
template <int DIN, int DOUT>
__global__ void node_mlp_wmma_kernel(const float* __restrict__ ego, const float* __restrict__ side,
                                     const float* __restrict__ W1, const float* __restrict__ b1,
                                     const float* __restrict__ W2, const float* __restrict__ b2,
                                     float* __restrict__ ego_next, int nTiles) {
    int wid = (blockIdx.x * blockDim.x + threadIdx.x) >> 5;
    if (wid >= nTiles) return;                 // wave-uniform: EXEC stays all-ones
    int lane = threadIdx.x & 31;
    int half = lane >> 4;
    int mrow = lane & 15;
    int m0 = wid * 16;
    int row = m0 + mrow;

    constexpr int NT = DOUT / 16;
    v8f acc1[NT] = {};
    v8f acc2[NT] = {};

    const float* egoRow  = ego  + (size_t)row * DIN;
    const float* sideRow = side + (size_t)row * DIN;

    for (int k = 0; k < DIN; k += 4) {
        int kk = k + 2 * half;
        float2 e2 = *(const float2*)(egoRow + kk);
        float2 s2 = *(const float2*)(sideRow + kk);
        v2f a1; a1.x = e2.x + s2.x; a1.y = e2.y + s2.y;   // (ego + side)
        v2f a2; a2.x = e2.x * s2.x; a2.y = e2.y * s2.y;   // (ego * side)
#pragma unroll
        for (int t = 0; t < NT; ++t) {
            int col = t * 16 + mrow;
            v2f bw1; bw1.x = W1[(size_t)kk * DOUT + col]; bw1.y = W1[(size_t)(kk + 1) * DOUT + col];
            v2f bw2; bw2.x = W2[(size_t)kk * DOUT + col]; bw2.y = W2[(size_t)(kk + 1) * DOUT + col];
            acc1[t] = __builtin_amdgcn_wmma_f32_16x16x4_f32(false, a1, false, bw1, (short)0,
                                                            acc1[t], false, false);
            acc2[t] = __builtin_amdgcn_wmma_f32_16x16x4_f32(false, a2, false, bw2, (short)0,
                                                            acc2[t], false, false);
        }
    }

#pragma unroll
    for (int t = 0; t < NT; ++t) {
        int col = t * 16 + mrow;
        float bb1 = b1[col], bb2 = b2[col];
#pragma unroll
        for (int i = 0; i < 8; ++i) {
            int r = m0 + i + 8 * half;
            float v1 = acc1[t][i] + bb1;
            v1 = v1 > 0.0f ? v1 : 0.01f * v1;
            float v2 = acc2[t][i] + bb2;
            v2 = v2 > 0.0f ? v2 : 0.01f * v2;
            ego_next[(size_t)r * DOUT + col] = v1 + v2;
        }
    }
}

// ------------------------------------------------- row-normalize into output
template <int DOUT>
__global__ void norm_out_kernel(const float* __restrict__ ego, float* __restrict__ out,
                                int colbase, int nNodes) {
    int n = blockIdx.x * blockDim.x + threadIdx.x;
    if (n >= nNodes) return;
    const float* rowp = ego + (size_t)n * DOUT;
    float ss = 0.0f;
#pragma unroll
    for (int j = 0; j < DOUT; ++j) { float v = rowp[j]; ss += v * v; }
    float nr = fmaxf(sqrtf(ss), 1e-12f);
    float inv = 1.0f / nr;
    float* o = out + (size_t)n * 240 + colbase;
#pragma unroll
    for (int j = 0; j < DOUT; ++j) o[j] = rowp[j] * inv;
}

// ---------------------------------------------------------------- launcher
extern "C" void kernel_launch(void* const* d_in, const int* in_sizes, int n_in,
                              void* d_out, int out_size, void* d_ws, size_t ws_size,
                              hipStream_t stream) {
    const float* ent = (const float*)d_in[0];
    const float* rel = (const float*)d_in[1];
    const float* W1p[3] = { (const float*)d_in[2], (const float*)d_in[6],  (const float*)d_in[10] };
    const float* b1p[3] = { (const float*)d_in[3], (const float*)d_in[7],  (const float*)d_in[11] };
    const float* W2p[3] = { (const float*)d_in[4], (const float*)d_in[8],  (const float*)d_in[12] };
    const float* b2p[3] = { (const float*)d_in[5], (const float*)d_in[9],  (const float*)d_in[13] };
    const int* esrc = (const int*)d_in[14];
    const int* edst = (const int*)d_in[15];
    const int* etyp = (const int*)d_in[16];
    float* out = (float*)d_out;

    const int N = NNODE, E = NEDGE;
    const float phase_scale = (float)(3.1415926235897933 / ((12.0 + 2.0) / 64.0));

    // workspace layout (floats)
    float* wsf  = (float*)d_ws;
    float* att  = wsf;                            // E
    float* segm = att + E;                        // N (stored as f32 bits, atomicMax on uint view)
    float* segs = segm + N;                       // N
    float* side = segs + N;                       // N*128 (reused per layer)
    float* ego1 = side + (size_t)N * 128;         // N*64
    float* ego2 = ego1 + (size_t)N * 64;          // N*32
    float* ego3 = ego2 + (size_t)N * 32;          // N*16

    // 1) zero segment max/sum, then raw attention + segment-max
    zero_kernel<<<(2 * N + 255) / 256, 256, 0, stream>>>(segm, 2LL * N);
    {
        long long threads = (long long)E * 32;
        edge_att_kernel<<<(unsigned)((threads + 255) / 256), 256, 0, stream>>>(
            ent, rel, esrc, edst, etyp, att, (unsigned int*)segm, phase_scale, E);
    }
    edge_expsum_kernel<<<(E + 255) / 256, 256, 0, stream>>>(segm, att, edst, segs, E);
    edge_norm_kernel<<<(E + 255) / 256, 256, 0, stream>>>(att, edst, segs, E);

    // 2) out[:, 0:128] = ent_embed
    {
        long long total = (long long)N * 128;
        copy_embed_kernel<<<(unsigned)((total + 255) / 256), 256, 0, stream>>>(ent, out, total);
    }

    const int nTiles = N / 16;  // 6250
    const long long eWaves = (long long)E * 32;
    const unsigned eGrid = (unsigned)((eWaves + 255) / 256);
    const unsigned gGrid = (unsigned)(((long long)nTiles * 32 + 127) / 128);

    // ---- layer 0: 128 -> 64
    zero_kernel<<<(unsigned)(((long long)N * 128 + 255) / 256), 256, 0, stream>>>(side, (long long)N * 128);
    scatter_kernel<128><<<eGrid, 256, 0, stream>>>(ent, att, esrc, edst, side, E);
    node_mlp_wmma_kernel<128, 64><<<gGrid, 128, 0, stream>>>(ent, side, W1p[0], b1p[0], W2p[0], b2p[0], ego1, nTiles);
    norm_out_kernel<64><<<(N + 255) / 256, 256, 0, stream>>>(ego1, out, 128, N);

    // ---- layer 1: 64 -> 32
    zero_kernel<<<(unsigned)(((long long)N * 64 + 255) / 256), 256, 0, stream>>>(side, (long long)N * 64);
    scatter_kernel<64><<<eGrid, 256, 0, stream>>>(ego1, att, esrc, edst, side, E);
    node_mlp_wmma_kernel<64, 32><<<gGrid, 128, 0, stream>>>(ego1, side, W1p[1], b1p[1], W2p[1], b2p[1], ego2, nTiles);
    norm_out_kernel<32><<<(N + 255) / 256, 256, 0, stream>>>(ego2, out, 192, N);

    // ---- layer 2: 32 -> 16
    zero_kernel<<<(unsigned)(((long long)N * 32 + 255) / 256), 256, 0, stream>>>(side, (long long)N * 32);
    scatter_kernel<32><<<eGrid, 256, 0, stream>>>(ego2, att, esrc, edst, side, E);
    node_mlp_wmma_kernel<32, 16><<<gGrid, 128, 0, stream>>>(ego2, side, W1p[2], b1p[2], W2p[2], b2p[2], ego3, nTiles);
    norm_out_kernel<16><<<(N + 255) / 256, 256, 0, stream>>>(ego3, out, 224, N);
}